// RobertaNerGatModel_85152021610659
// MI455X (gfx1250) — compile-verified
//
#include <hip/hip_runtime.h>
#include <math.h>

#define Bdim 8
#define Sdim 512
#define Ddim 768
#define Hh   12
#define Cc   64
#define HC   768
#define Ee   131072
#define NCls 9
#define Nn   (Bdim * Sdim)      /* 4096 nodes */
#define XS   772                /* padded x row stride (multiple of 4) */
#define EN   (Ee + Nn)          /* edges incl. self loops = 135168 */

typedef __attribute__((ext_vector_type(2))) float v2f;
typedef __attribute__((ext_vector_type(8))) float v8f;

// ---------------------------------------------------------------- init small
__global__ void k_init_small(float* __restrict__ denom, float* __restrict__ msum) {
    int t = blockIdx.x * blockDim.x + threadIdx.x;
    if (t < Nn * Hh) denom[t] = 0.0f;
    if (t == 0) msum[0] = 0.0f;
}

// ---------------------------------------------------------------- build x = [seq | sigmoid(z)], padded stride XS
__global__ void k_prep_x(const float* __restrict__ seq, const float* __restrict__ z,
                         float* __restrict__ x) {
    int t = blockIdx.x * blockDim.x + threadIdx.x;     // over Nn * 769
    if (t >= Nn * (Ddim + 1)) return;
    int n = t / (Ddim + 1);
    int d = t - n * (Ddim + 1);
    float v;
    if (d < Ddim) v = seq[(size_t)n * Ddim + d];
    else          v = 1.0f / (1.0f + __expf(-z[n]));   // sigmoid
    x[(size_t)n * XS + d] = v;
}

// ---------------------------------------------------------------- sum(edge_attr) for self-loop fill value
__global__ void k_edge_sum(const float* __restrict__ ea, float* __restrict__ msum) {
    __shared__ float sm[256];
    float s = 0.0f;
    for (int i = blockIdx.x * blockDim.x + threadIdx.x; i < Ee; i += gridDim.x * blockDim.x)
        s += ea[i];
    sm[threadIdx.x] = s;
    __syncthreads();
    for (int o = 128; o > 0; o >>= 1) {
        if (threadIdx.x < o) sm[threadIdx.x] += sm[threadIdx.x + o];
        __syncthreads();
    }
    if (threadIdx.x == 0) atomicAdd(msum, sm[0]);
}

// ---------------------------------------------------------------- k[h] = sum_c W_edge[h*64+c] * att_edge[h*64+c]
__global__ void k_kedge(const float* __restrict__ Wedge, const float* __restrict__ att_e,
                        float* __restrict__ kedge) {
    int h = threadIdx.x;
    if (h >= Hh) return;
    float s = 0.0f;
    #pragma unroll 8
    for (int c = 0; c < Cc; ++c) s += Wedge[h * Cc + c] * att_e[h * Cc + c];
    kedge[h] = s;
}

// ---------------------------------------------------------------- hx = x[:, :768] @ W_lin[:768,:] + sigmoid(z) ⊗ W_lin[768,:]
// fp32 WMMA 16x16x4, one 16x16 output tile per wave, K loop of 192 steps.
__global__ __launch_bounds__(128) void k_gemm_wmma(const float* __restrict__ x,
                                                   const float* __restrict__ W,
                                                   float* __restrict__ hx) {
    const int lane = threadIdx.x & 31;
    const int wave = threadIdx.x >> 5;
    const int mT   = blockIdx.x;                 // 0..255
    const int nT   = blockIdx.y * 4 + wave;      // 0..47
    const int half = lane >> 4;                  // 0: K{0,1}, 1: K{2,3}
    const int l15  = lane & 15;
    const int row  = mT * 16 + l15;              // A: M = lane&15 in both halves
    const int col  = nT * 16 + l15;              // B/C/D: N = lane&15
    const float* Arow = x + (size_t)row * XS + (half << 1);     // 8B aligned (XS mult of 4)
    const float* Bcol = W + (size_t)(half << 1) * HC + col;
    v8f c = {0.f, 0.f, 0.f, 0.f, 0.f, 0.f, 0.f, 0.f};
    for (int k0 = 0; k0 < Ddim; k0 += 4) {
        v2f a = *(const v2f*)(Arow + k0);
        v2f b;
        b.x = Bcol[(size_t)k0 * HC];
        b.y = Bcol[(size_t)(k0 + 1) * HC];
        c = __builtin_amdgcn_wmma_f32_16x16x4_f32(
                /*neg_a=*/false, a, /*neg_b=*/false, b,
                /*c_mod=*/(short)0, c, /*reuse_a=*/false, /*reuse_b=*/false);
    }
    // rank-1 update for the sigmoid(z) column (W_lin row 768), then store
    const float wz = W[(size_t)Ddim * HC + col];
    #pragma unroll
    for (int r = 0; r < 8; ++r) {
        int m = mT * 16 + (half << 3) + r;       // C/D: M = r + 8*half
        float v = c[r] + x[(size_t)m * XS + Ddim] * wz;
        hx[(size_t)m * HC + col] = v;
    }
}

// ---------------------------------------------------------------- zero accumulator
__global__ void k_zero(float* __restrict__ p, int n) {
    int t = blockIdx.x * blockDim.x + threadIdx.x;
    if (t < n) p[t] = 0.0f;
}

// ---------------------------------------------------------------- a_src[n,h], a_dst[n,h]
__global__ void k_attn_coef(const float* __restrict__ hx, const float* __restrict__ att_s,
                            const float* __restrict__ att_d, float* __restrict__ asrc,
                            float* __restrict__ adst) {
    int t = blockIdx.x * blockDim.x + threadIdx.x;   // over Nn*Hh
    if (t >= Nn * Hh) return;
    int n = t / Hh, h = t - n * Hh;
    const float* hrow = hx + (size_t)n * HC + h * Cc;
    float s0 = 0.0f, s1 = 0.0f;
    #pragma unroll 8
    for (int c = 0; c < Cc; ++c) {
        float v = hrow[c];
        s0 += v * att_s[h * Cc + c];
        s1 += v * att_d[h * Cc + c];
    }
    asrc[t] = s0;
    adst[t] = s1;
}

// ---------------------------------------------------------------- per-edge logits -> exp, accumulate denom
__global__ void k_edge_softmax(const int* __restrict__ ei, const float* __restrict__ ea,
                               const float* __restrict__ asrc, const float* __restrict__ adst,
                               const float* __restrict__ kedge, const float* __restrict__ msum,
                               float* __restrict__ exw, float* __restrict__ denom) {
    int t = blockIdx.x * blockDim.x + threadIdx.x;   // over EN*Hh
    if (t >= EN * Hh) return;
    int e = t / Hh, h = t - e * Hh;
    int s, d;
    float eav;
    if (e < Ee) { s = ei[e]; d = ei[Ee + e]; eav = ea[e]; }
    else        { s = e - Ee; d = s;         eav = msum[0] * (1.0f / (float)Ee); }
    float a = asrc[s * Hh + h] + adst[d * Hh + h] + eav * kedge[h];
    a = (a > 0.0f) ? a : 0.2f * a;                    // LeakyReLU(0.2)
    float ex = __expf(a);                             // max-shift cancels in the quotient
    exw[t] = ex;
    atomicAdd(&denom[d * Hh + h], ex);
}

// ---------------------------------------------------------------- normalize alpha once per (e,h)
__global__ void k_norm(const int* __restrict__ ei, const float* __restrict__ denom,
                       float* __restrict__ exw) {
    int t = blockIdx.x * blockDim.x + threadIdx.x;   // over EN*Hh
    if (t >= EN * Hh) return;
    int e = t / Hh, h = t - e * Hh;
    int d = (e < Ee) ? ei[Ee + e] : (e - Ee);
    exw[t] = exw[t] / (denom[d * Hh + h] + 1e-16f);
}

// ---------------------------------------------------------------- message scatter: out[dst] += alpha * hx[src]
__global__ void k_scatter(const int* __restrict__ ei, const float* __restrict__ hx,
                          const float* __restrict__ exw, float* __restrict__ out) {
    long long t = (long long)blockIdx.x * blockDim.x + threadIdx.x;   // over EN*HC
    if (t >= (long long)EN * HC) return;
    int e = (int)(t / HC);
    int j = (int)(t - (long long)e * HC);
    int h = j >> 6;
    int s, d;
    if (e < Ee) { s = ei[e]; d = ei[Ee + e]; }
    else        { s = e - Ee; d = s; }
    float w = exw[e * Hh + h];
    atomicAdd(&out[(size_t)d * HC + j], w * hx[(size_t)s * HC + j]);
}

// ---------------------------------------------------------------- g = elu(out + bias), in place
__global__ void k_elu_bias(float* __restrict__ out, const float* __restrict__ bias) {
    int t = blockIdx.x * blockDim.x + threadIdx.x;   // over Nn*HC
    if (t >= Nn * HC) return;
    float v = out[t] + bias[t % HC];
    out[t] = (v > 0.0f) ? v : expm1f(v);             // ELU alpha=1
}

// ---------------------------------------------------------------- logits = g @ cls_W + cls_b
__global__ void k_cls(const float* __restrict__ g, const float* __restrict__ W,
                      const float* __restrict__ b, float* __restrict__ logits) {
    int t = blockIdx.x * blockDim.x + threadIdx.x;   // over Nn*NCls
    if (t >= Nn * NCls) return;
    int n = t / NCls, cl = t - n * NCls;
    const float* grow = g + (size_t)n * HC;
    float acc = b[cl];
    #pragma unroll 8
    for (int j = 0; j < HC; ++j) acc += grow[j] * W[j * NCls + cl];
    logits[t] = acc;
}

// ================================================================ host
extern "C" void kernel_launch(void* const* d_in, const int* in_sizes, int n_in,
                              void* d_out, int out_size, void* d_ws, size_t ws_size,
                              hipStream_t stream) {
    const float* seq    = (const float*)d_in[0];   // [B,S,D]
    const float* zsc    = (const float*)d_in[1];   // [B,S]
    const int*   ei     = (const int*)  d_in[2];   // [2,E]
    const float* eattr  = (const float*)d_in[3];   // [E,1]
    const float* Wlin   = (const float*)d_in[4];   // [D+1, HC]
    const float* Wedge  = (const float*)d_in[5];   // [1, HC]
    const float* att_s  = (const float*)d_in[6];   // [H,C]
    const float* att_d  = (const float*)d_in[7];   // [H,C]
    const float* att_e  = (const float*)d_in[8];   // [H,C]
    const float* gbias  = (const float*)d_in[9];   // [HC]
    const float* clsW   = (const float*)d_in[10];  // [HC, NC]
    const float* clsB   = (const float*)d_in[11];  // [NC]
    float* logits = (float*)d_out;                 // [B,S,NC]

    // workspace layout (floats); x-buffer is reused as the scatter accumulator
    float* ws    = (float*)d_ws;
    float* xb    = ws;                               // Nn*XS   (alias: outacc, Nn*HC <= Nn*XS)
    float* hx    = xb    + (size_t)Nn * XS;          // Nn*HC
    float* asrc  = hx    + (size_t)Nn * HC;          // Nn*Hh
    float* adst  = asrc  + (size_t)Nn * Hh;          // Nn*Hh
    float* denom = adst  + (size_t)Nn * Hh;          // Nn*Hh
    float* kedge = denom + (size_t)Nn * Hh;          // 16
    float* msum  = kedge + 16;                       // 4
    float* exw   = msum  + 4;                        // EN*Hh
    float* outacc = xb;                              // alias (safe: x dead after GEMM)

    const int T = 256;
    // 1) zero denom + edge-attr sum slot
    k_init_small<<<(Nn * Hh + T - 1) / T, T, 0, stream>>>(denom, msum);
    // 2) build x (padded stride)
    k_prep_x<<<(Nn * (Ddim + 1) + T - 1) / T, T, 0, stream>>>(seq, zsc, xb);
    // 3) sum(edge_attr)
    k_edge_sum<<<512, T, 0, stream>>>(eattr, msum);
    // 4) k[h] scalars
    k_kedge<<<1, 32, 0, stream>>>(Wedge, att_e, kedge);
    // 5) hx = x @ W_lin via fp32 WMMA  (grid: 256 M-tiles x 12 groups of 4 N-tiles)
    k_gemm_wmma<<<dim3(Nn / 16, (HC / 16) / 4), 128, 0, stream>>>(xb, Wlin, hx);
    // 6) zero scatter accumulator (aliases x, so must follow GEMM)
    k_zero<<<(Nn * HC + T - 1) / T, T, 0, stream>>>(outacc, Nn * HC);
    // 7) a_src / a_dst
    k_attn_coef<<<(Nn * Hh + T - 1) / T, T, 0, stream>>>(hx, att_s, att_d, asrc, adst);
    // 8) exp(leaky(logit)) + denom
    k_edge_softmax<<<(EN * Hh + T - 1) / T, T, 0, stream>>>(ei, eattr, asrc, adst,
                                                            kedge, msum, exw, denom);
    // 9) alpha = ex / denom[dst]
    k_norm<<<(EN * Hh + T - 1) / T, T, 0, stream>>>(ei, denom, exw);
    // 10) out[dst] += alpha * hx[src]
    long long scat = (long long)EN * HC;
    k_scatter<<<(unsigned)((scat + T - 1) / T), T, 0, stream>>>(ei, hx, exw, outacc);
    // 11) elu(out + bias)
    k_elu_bias<<<(Nn * HC + T - 1) / T, T, 0, stream>>>(outacc, gbias);
    // 12) classifier
    k_cls<<<(Nn * NCls + T - 1) / T, T, 0, stream>>>(outacc, clsW, clsB, logits);

    (void)in_sizes; (void)n_in; (void)out_size; (void)ws_size;
}